// GeoDCD_87651692576905
// MI455X (gfx1250) — compile-verified
//
#include <hip/hip_runtime.h>
#include <math.h>

// ---------------------------------------------------------------------------
// GeoDCD: out[b,n,t,c] = tanh( sum_k z[b, nbr[n,k], t, c] * EW[c,n,k] )
//   EW[c,n,k] = (sum_base CC[c,base]*BW[base,n,k]) * adj[n,k]
//
// MI455X reasoning:
//  * Memory-bound target: ~1.07 GFLOP vs 256 MiB compulsory HBM traffic
//    (z read once, out written once) -> ~11us floor at 23.3 TB/s. The 2 GiB
//    gather amplification is absorbed by the 192 MB L2 (z = 128 MiB fits).
//  * fp32 throughout; non-temporal output stores keep z resident in L2.
//  * WMMA where it fits exactly: per-node EW = CC[64x4] x (adj*BW)[4x16] via
//    V_WMMA_F32_16X16X4_F32, one full 16x16 c-tile per wave.
//  * Neighbor indices are block-uniform: readfirstlane -> SGPR base math.
//    Per-k row bases for BOTH batches are pinned into SGPR pairs with an
//    empty asm barrier (so all 64 gathers use saddr + shared 32-bit voffset
//    and the reassociator can't pull the +64MiB batch offset into VALU),
//    carried as explicit address_space(1) pointers so they stay
//    GLOBAL_LOAD_B128 (LOADcnt only) instead of decaying to FLAT
//    (LOADcnt+DScnt, which would serialize the LDS weight broadcasts).
//  * tanh via v_exp_f32 + v_rcp_f32 (2 trans + ~6 VALU/elem) instead of
//    ocml tanh (~28 VALU/elem): 33.5M tanh evaluations were the dominant
//    VALU cost and would otherwise exceed the HBM roofline.
// ---------------------------------------------------------------------------

typedef float v2f __attribute__((ext_vector_type(2)));
typedef float v4f __attribute__((ext_vector_type(4)));
typedef float v8f __attribute__((ext_vector_type(8)));

typedef const float __attribute__((address_space(1)))* gfp;   // global float*
typedef const v4f   __attribute__((address_space(1)))* gv4p;  // global v4f*

#define B_  2
#define N_  8192
#define T_  32
#define C_  64
#define K_  16
#define NB_ 4

__device__ __forceinline__ float fast_tanh(float x) {
    // tanh(x) = sign(x) * (1-q)/(1+q),  q = exp2(-2*log2(e)*|x|)
    const float NEG2LOG2E = -2.8853900817779268f;
    float a = __builtin_fabsf(x);                       // free src modifier
    float q = __builtin_amdgcn_exp2f(NEG2LOG2E * a);    // v_exp_f32
    float r = (1.0f - q) * __builtin_amdgcn_rcpf(1.0f + q); // v_rcp_f32
    return __builtin_copysignf(r, x);
}

__device__ __forceinline__ v4f tanh4(v4f x) {
    v4f r;
    r.x = fast_tanh(x.x);
    r.y = fast_tanh(x.y);
    r.z = fast_tanh(x.z);
    r.w = fast_tanh(x.w);
    return r;
}

__global__ __launch_bounds__(256)
void geodcd_kernel(const float* __restrict__ z,     // [B,N,T,C]
                   const int*   __restrict__ nbr,   // [N,K]
                   const float* __restrict__ adj,   // [N,K]
                   const float* __restrict__ bw,    // [NB,N,K]
                   const float* __restrict__ cc,    // [C,NB]
                   float*       __restrict__ out)   // [B,N,T,C]
{
    __shared__ float s_ew[K_ * C_];   // EW stored [k][c] for broadcast reads

    const int n    = blockIdx.x;
    const int tid  = threadIdx.x;
    const int wave = tid >> 5;
    const int lane = tid & 31;
    const int hi   = lane >> 4;       // lane half (WMMA layout selector)
    const int l16  = lane & 15;

    // --- Waves 0..3: EW[c_tile(16), k(16)] = CC_tile[16x4] x (adj*BW)[4x16] ---
    if (wave < 4) {
        const int ct = wave;                       // c-tile index
        // A (16x4 f32): lanes 0-15 -> {K=0,K=1}, lanes 16-31 -> {K=2,K=3}
        const int crow = ct * 16 + l16;
        v2f a;
        a.x = cc[crow * NB_ + 2 * hi + 0];
        a.y = cc[crow * NB_ + 2 * hi + 1];
        // B (4x16 f32): row = base, col = neighbor k; same K striping as A
        const int k  = l16;
        const float ad = adj[n * K_ + k];
        v2f b;
        b.x = ad * bw[(2 * hi + 0) * (N_ * K_) + n * K_ + k];
        b.y = ad * bw[(2 * hi + 1) * (N_ * K_) + n * K_ + k];

        v8f d = {};
        // 8 args: (neg_a, A, neg_b, B, c_mod, C, reuse_a, reuse_b)
        d = __builtin_amdgcn_wmma_f32_16x16x4_f32(
                false, a, false, b, (short)0, d, false, false);

        // D (16x16 f32): lane holds column N=l16, rows v + 8*hi
        #pragma unroll
        for (int v = 0; v < 8; ++v) {
            const int cr = ct * 16 + v + 8 * hi;
            s_ew[k * C_ + cr] = d[v];
        }
    }

    // --- Neighbor indices: block-uniform -> keep scalar (SGPR) ---
    int idxs[K_];
    #pragma unroll
    for (int k = 0; k < K_; ++k)
        idxs[k] = __builtin_amdgcn_readfirstlane(nbr[n * K_ + k]);

    __syncthreads();

    // --- Gather + accumulate. thread = (c-quad, t); covers both b and 2 t's ---
    const int c4 = tid & 15;          // which float4 of the 64 channels
    const int tq = tid >> 4;          // 0..15
    const int t0 = tq, t1 = tq + 16;
    const int co = c4 * 4;

    // Per-thread 32-bit element offsets, shared across all k and both batches.
    const unsigned off00 = (unsigned)(t0 * C_ + co);
    const unsigned off01 = (unsigned)(t1 * C_ + co);   // = off00 + 0x1000 bytes

    const size_t bstride = (size_t)N_ * T_ * C_;       // batch stride (elements)
    const float* __restrict__ zb1 = z + bstride;       // uniform batch-1 base

    v4f acc00 = {}, acc01 = {}, acc10 = {}, acc11 = {};

    #pragma unroll
    for (int k = 0; k < K_; ++k) {
        const size_t nodeoff = (size_t)((unsigned)idxs[k]) * (T_ * C_); // SALU
        gfp r0 = (gfp)(z   + nodeoff);   // batch-0 row base (global addrspace)
        gfp r1 = (gfp)(zb1 + nodeoff);   // batch-1 row base (global addrspace)
        // Pin row bases into SGPR pairs; opaque to the reassociator so the
        // +bstride offset cannot be pulled back into per-lane VALU adds.
        // Address-space(1) type keeps them GLOBAL_LOAD (LOADcnt-only).
        asm("" : "+s"(r0), "+s"(r1));
        const v4f z00 = *(gv4p)(r0 + off00);
        const v4f z01 = *(gv4p)(r0 + off01);
        const v4f z10 = *(gv4p)(r1 + off00);
        const v4f z11 = *(gv4p)(r1 + off01);
        const v4f w   = *(const v4f*)(s_ew + k * C_ + co);   // LDS broadcast
        acc00 += z00 * w;
        acc01 += z01 * w;
        acc10 += z10 * w;
        acc11 += z11 * w;
    }

    // --- tanh + non-temporal stores (don't evict z from L2) ---
    const size_t o00 = ((size_t)n * T_ + t0) * C_ + co;
    const size_t o01 = ((size_t)n * T_ + t1) * C_ + co;
    __builtin_nontemporal_store(tanh4(acc00), (v4f*)(out + o00));
    __builtin_nontemporal_store(tanh4(acc01), (v4f*)(out + o01));
    __builtin_nontemporal_store(tanh4(acc10), (v4f*)(out + bstride + o00));
    __builtin_nontemporal_store(tanh4(acc11), (v4f*)(out + bstride + o01));
}

extern "C" void kernel_launch(void* const* d_in, const int* in_sizes, int n_in,
                              void* d_out, int out_size, void* d_ws, size_t ws_size,
                              hipStream_t stream) {
    const float* z   = (const float*)d_in[0];  // [B,N,T,C]
    const int*   idx = (const int*)  d_in[1];  // [N,K]
    const float* adj = (const float*)d_in[2];  // [N,K]
    const float* bw  = (const float*)d_in[3];  // [NB,N,K]
    const float* cc  = (const float*)d_in[4];  // [C,NB]
    float* out = (float*)d_out;                // [B,N,T,C]

    geodcd_kernel<<<N_, 256, 0, stream>>>(z, idx, adj, bw, cc, out);
}